// loss_concentration_42674795053844
// MI455X (gfx1250) — compile-verified
//
#include <hip/hip_runtime.h>

// Problem constants from the reference: B=32, K=64, H=W=128.
#define HW_H 128
#define HW_W 128
#define TPAD 132              // LDS row stride in floats (128 + 4 pad; keeps 16B align)

typedef int v4i __attribute__((vector_size(16)));           // matches builtin param pointee
typedef __attribute__((address_space(1))) v4i gbuf_t;       // global 16B chunk
typedef __attribute__((address_space(3))) v4i lbuf_t;       // LDS 16B chunk
typedef __attribute__((address_space(3))) float lfloat;

__device__ __forceinline__ void async_cp_b128(const float* g, float* l) {
#if __has_builtin(__builtin_amdgcn_global_load_async_to_lds_b128)
  __builtin_amdgcn_global_load_async_to_lds_b128((gbuf_t*)g, (lbuf_t*)l, 0, 0);
#else
  unsigned lds_addr = (unsigned)(unsigned long long)(lfloat*)l;
  asm volatile("global_load_async_to_lds_b128 %0, %1, off"
               :: "v"(lds_addr), "v"(g) : "memory");
#endif
}

__device__ __forceinline__ void wait_async0() {
#if __has_builtin(__builtin_amdgcn_s_wait_asynccnt)
  __builtin_amdgcn_s_wait_asynccnt(0);
#else
  asm volatile("s_wait_asynccnt 0" ::: "memory");
#endif
}

__global__ __launch_bounds__(256)
void conc_partial_kernel(const float* __restrict__ D,
                         const float* __restrict__ zeta,
                         float* __restrict__ partial) {
  __shared__ float tile[64 * TPAD];                  // 33,792 B
  __shared__ float redA[64], redB[64], redC[128], redD[128];

  const int t  = threadIdx.x;
  const int bk = blockIdx.x;
  const float* base = D + (size_t)bk * (HW_H * HW_W);

  const float invN   = 1.0f / 128.0f;
  const float invNm1 = 1.0f / 127.0f;

  float vxSum = 0.0f, vxSq = 0.0f;   // row-thread accumulators (over rows)
  float cS = 0.0f, cQ = 0.0f;        // col-thread accumulators (over 128 rows)

  for (int half = 0; half < 2; ++half) {
    // ---- async global -> LDS: 64 rows x 128 cols (8 iters x 256 lanes x 16B) ----
    {
      const int wv  = t >> 5;          // wave id 0..7 -> row within group of 8
      const int col = (t & 31) * 4;    // 4 floats per lane
      #pragma unroll
      for (int i = 0; i < 8; ++i) {
        const int rl = i * 8 + wv;     // local row 0..63
        const float* g = base + (size_t)(half * 64 + rl) * HW_W + col;
        async_cp_b128(g, &tile[rl * TPAD + col]);
      }
      wait_async0();
    }
    __syncthreads();

    if (t < 64) {
      // row statistics for local row t (global row half*64 + t)
      const float4* rp = reinterpret_cast<const float4*>(&tile[t * TPAD]);
      float S = 0.0f, Q = 0.0f;
      #pragma unroll
      for (int i = 0; i < 32; ++i) {
        float4 v = rp[i];
        S += v.x + v.y + v.z + v.w;
        Q += v.x * v.x + v.y * v.y + v.z * v.z + v.w * v.w;
      }
      const float vx = (Q - S * S * invN) * invNm1;  // unbiased var over W
      vxSum += vx;
      vxSq  += vx * vx;
    } else if (t < 192) {
      // column statistics for column t-64, accumulated across both halves
      const int c = t - 64;
      #pragma unroll 8
      for (int r = 0; r < 64; ++r) {
        const float v = tile[r * TPAD + c];
        cS += v;
        cQ += v * v;
      }
    }
    __syncthreads();   // tile may be overwritten by next half's async copy
  }

  if (t < 64) {
    redA[t] = vxSum;
    redB[t] = vxSq;
  } else if (t < 192) {
    const float vy = (cQ - cS * cS * invN) * invNm1; // unbiased var over H
    redC[t - 64] = vy;
    redD[t - 64] = vy * vy;
  }
  __syncthreads();

  // deterministic tree reductions
  for (int s = 32; s > 0; s >>= 1) {
    if (t < s) { redA[t] += redA[t + s]; redB[t] += redB[t + s]; }
    __syncthreads();
  }
  for (int s = 64; s > 0; s >>= 1) {
    if (t < s) { redC[t] += redC[t + s]; redD[t] += redD[t + s]; }
    __syncthreads();
  }

  if (t == 0) {
    const float stdx_D = (redB[0] - redA[0] * redA[0] * invN) * invNm1;
    const float stdy_D = (redD[0] - redC[0] * redC[0] * invN) * invNm1;
    const float z  = zeta[bk];
    const float z2 = z * z;
    const float inv4 = 1.0f / (z2 * z2);             // g = D/z  =>  var(var) scales 1/z^4
    partial[bk] = 6.28318530717958647692f * expf(stdx_D * inv4) + expf(stdy_D * inv4);
  }
}

__global__ __launch_bounds__(256)
void conc_finalize_kernel(const float* __restrict__ partial,
                          float* __restrict__ out, int n) {
  __shared__ float red[256];
  const int t = threadIdx.x;
  float s = 0.0f;
  for (int i = t; i < n; i += 256) s += partial[i];  // fixed order per thread
  red[t] = s;
  __syncthreads();
  for (int k = 128; k > 0; k >>= 1) {
    if (t < k) red[t] += red[t + k];
    __syncthreads();
  }
  if (t == 0) out[0] = red[0];
}

extern "C" void kernel_launch(void* const* d_in, const int* in_sizes, int n_in,
                              void* d_out, int out_size, void* d_ws, size_t ws_size,
                              hipStream_t stream) {
  const float* D    = (const float*)d_in[0];   // [B,K,H,W] f32
  const float* zeta = (const float*)d_in[1];   // [B,K]     f32
  float* ws  = (float*)d_ws;                   // nbk partials
  float* out = (float*)d_out;                  // scalar
  const int nbk = in_sizes[1];                 // B*K = 2048

  conc_partial_kernel<<<nbk, 256, 0, stream>>>(D, zeta, ws);
  conc_finalize_kernel<<<1, 256, 0, stream>>>(ws, out, nbk);
}